// RGCN_50938312130617
// MI455X (gfx1250) — compile-verified
//
#include <hip/hip_runtime.h>
#include <stdint.h>

typedef __attribute__((ext_vector_type(2))) float v2f;
typedef __attribute__((ext_vector_type(8))) float v8f;

#define DD 128
#define RR 4
#define XS_STRIDE 132  // pad 16x128 fp32 tile -> conflict-free ds_load_b64
                       // row pitch 528 B (16B aligned, ok for async B128)

// ---------------------------------------------------------------------------
// GEMM: for one 16-row node tile, compute h_r = X @ W_r (r=0..3) and
// out = X @ root + bias (mat==4). grid = (ceil(N/16), 5), block = 256 (8 waves)
// X tile staged memory->LDS with GLOBAL_LOAD_ASYNC_TO_LDS_B128 (ASYNCcnt),
// then 32 x v_wmma_f32_16x16x4_f32 per wave over K=128.
// ---------------------------------------------------------------------------
__global__ __launch_bounds__(256) void rgcn_gemm_wmma(
    const float* __restrict__ x,      // [N, D]
    const float* __restrict__ wrel,   // [R, D, D] (this layer)
    const float* __restrict__ wroot,  // [D, D]
    const float* __restrict__ bias,   // [D]
    float* __restrict__ hbuf,         // [R, N, D]
    float* __restrict__ outbuf,       // [N, D]
    int N)
{
    __shared__ float Xs[16 * XS_STRIDE];

    const int tid  = threadIdx.x;
    const int row0 = blockIdx.x * 16;
    const int mat  = blockIdx.y;      // 0..3 = relation, 4 = root

    // ---- Stage the 16x128 X tile directly into LDS (async, no VGPR bounce).
    // 512 B128 chunks (16 rows x 32 chunks), 2 per thread.
    // LDS byte address = low 32 bits of the generic pointer (aperture rule).
    const uint32_t ldsBase = (uint32_t)(uintptr_t)&Xs[0];
    for (int c = tid; c < 16 * 32; c += 256) {
        int rr = c >> 5;           // row in tile
        int cc = (c & 31) * 4;     // float column (16B chunk)
        int gr = row0 + rr;
        uint32_t laddr = ldsBase + (uint32_t)(rr * XS_STRIDE + cc) * 4u;
        if (gr < N) {
            uint64_t gaddr = (uint64_t)(uintptr_t)(x + (size_t)gr * DD + cc);
            asm volatile("global_load_async_to_lds_b128 %0, %1, off"
                         :: "v"(laddr), "v"(gaddr)
                         : "memory");
        } else {
            float4 z = {0.f, 0.f, 0.f, 0.f};
            *(float4*)&Xs[rr * XS_STRIDE + cc] = z;
        }
    }
    asm volatile("s_wait_asynccnt 0" ::: "memory");
    __syncthreads();

    const float* __restrict__ W = (mat < RR) ? (wrel + (size_t)mat * DD * DD)
                                             : wroot;

    const int lane = tid & 31;
    const int wave = tid >> 5;
    const int j0   = wave * 16;      // output-column tile base
    const int l15  = lane & 15;
    const int half = lane >> 4;      // 0: K=k0,k0+1   1: K=k0+2,k0+3
    const int koff = half * 2;

    v8f acc = {};
    #pragma unroll 4
    for (int k0 = 0; k0 < DD; k0 += 4) {
        // A fragment (16x4 f32): lane half selects K pair, row = l15 (ds_load_b64)
        v2f a, b;
        const float* ap = &Xs[l15 * XS_STRIDE + k0 + koff];
        a.x = ap[0];
        a.y = ap[1];
        // B fragment (4x16 f32): row K = k0+koff(+1), col = j0 + l15
        b.x = W[(size_t)(k0 + koff) * DD + j0 + l15];
        b.y = W[(size_t)(k0 + koff + 1) * DD + j0 + l15];
        if (k0 + 4 < DD) {
            __builtin_prefetch(&W[(size_t)(k0 + 4 + koff) * DD + j0 + l15], 0, 1);
        }
        // D = A*B + C  (v_wmma_f32_16x16x4_f32)
        acc = __builtin_amdgcn_wmma_f32_16x16x4_f32(
            /*neg_a=*/false, a, /*neg_b=*/false, b,
            /*c_mod=*/(short)0, acc, /*reuse_a=*/false, /*reuse_b=*/false);
    }

    // C/D layout: lanes 0-15 -> N=l15, M=v ; lanes 16-31 -> N=l15, M=8+v
    float bv = 0.0f;
    float* __restrict__ dstp;
    if (mat < RR) {
        dstp = hbuf + (size_t)mat * N * DD;
    } else {
        dstp = outbuf;
        bv = bias[j0 + l15];
    }
    const int mrow = half * 8;
    #pragma unroll
    for (int v = 0; v < 8; ++v) {
        int gr = row0 + mrow + v;
        if (gr < N) dstp[(size_t)gr * DD + j0 + l15] = acc[v] + bv;
    }
}

// ---------------------------------------------------------------------------
__global__ void zero_f32(float* __restrict__ p, int n) {
    int i = blockIdx.x * blockDim.x + threadIdx.x;
    for (; i < n; i += gridDim.x * blockDim.x) p[i] = 0.0f;
}

// Per-(relation, dst) in-degree counts (fp32, matches reference semantics).
__global__ void rgcn_count(const int* __restrict__ dst,
                           const int* __restrict__ et,
                           float* __restrict__ cnt, int E, int N) {
    int e = blockIdx.x * blockDim.x + threadIdx.x;
    if (e < E) atomicAdd(&cnt[(size_t)et[e] * N + dst[e]], 1.0f);
}

// Fused scatter-mean: one wave per edge, one float4 per lane (32*4 = 128).
// out[dst] += h[etype][src] * (1 / max(cnt[etype][dst], 1))
__global__ void rgcn_scatter(const int* __restrict__ src,
                             const int* __restrict__ dsti,
                             const int* __restrict__ et,
                             const float* __restrict__ cnt,
                             const float* __restrict__ hbuf,
                             float* __restrict__ outbuf, int E, int N) {
    long long gid = (long long)blockIdx.x * blockDim.x + threadIdx.x;
    int e = (int)(gid >> 5);
    int q = ((int)gid & 31) * 4;
    if (e >= E) return;
    int r = et[e];
    int s = src[e];
    int d = dsti[e];
    float c = cnt[(size_t)r * N + d];
    float inv = 1.0f / fmaxf(c, 1.0f);
    const float4 h = *(const float4*)&hbuf[((size_t)r * N + s) * DD + q];
    float* o = &outbuf[(size_t)d * DD + q];
    atomicAdd(o + 0, h.x * inv);
    atomicAdd(o + 1, h.y * inv);
    atomicAdd(o + 2, h.z * inv);
    atomicAdd(o + 3, h.w * inv);
}

__global__ void relu_f32(float* __restrict__ p, int n) {
    int i = blockIdx.x * blockDim.x + threadIdx.x;
    for (; i < n; i += gridDim.x * blockDim.x) p[i] = fmaxf(p[i], 0.0f);
}

// ---------------------------------------------------------------------------
extern "C" void kernel_launch(void* const* d_in, const int* in_sizes, int n_in,
                              void* d_out, int out_size, void* d_ws, size_t ws_size,
                              hipStream_t stream) {
    const float* x       = (const float*)d_in[0];  // [N, D]
    const int*   eidx    = (const int*)d_in[1];    // [2, E] row-major: src then dst
    const int*   etype   = (const int*)d_in[2];    // [E]
    const float* weights = (const float*)d_in[3];  // [L, R, D, D]
    const float* roots   = (const float*)d_in[4];  // [L, D, D]
    const float* biases  = (const float*)d_in[5];  // [L, D]
    float* out = (float*)d_out;

    const int N = in_sizes[0] / DD;
    const int E = in_sizes[2];
    const int* srcI = eidx;
    const int* dstI = eidx + E;

    float* ws   = (float*)d_ws;
    float* hbuf = ws;                                 // [R, N, D]
    float* xbuf = hbuf + (size_t)RR * N * DD;         // [N, D]  layer-0 output
    float* cnt  = xbuf + (size_t)N * DD;              // [R, N]

    const int ntiles = (N + 15) / 16;
    dim3 ggrid(ntiles, RR + 1);
    const int cntN = RR * N;
    const long long scat_threads = (long long)E * 32;
    const int scat_blocks = (int)((scat_threads + 255) / 256);

    // ----- Layer 0: x -> xbuf -----
    zero_f32<<<(cntN + 255) / 256, 256, 0, stream>>>(cnt, cntN);
    rgcn_gemm_wmma<<<ggrid, 256, 0, stream>>>(
        x, weights, roots, biases, hbuf, xbuf, N);
    rgcn_count<<<(E + 255) / 256, 256, 0, stream>>>(dstI, etype, cnt, E, N);
    rgcn_scatter<<<scat_blocks, 256, 0, stream>>>(
        srcI, dstI, etype, cnt, hbuf, xbuf, E, N);
    relu_f32<<<1024, 256, 0, stream>>>(xbuf, N * DD);

    // ----- Layer 1: xbuf -> d_out -----
    zero_f32<<<(cntN + 255) / 256, 256, 0, stream>>>(cnt, cntN);
    rgcn_gemm_wmma<<<ggrid, 256, 0, stream>>>(
        xbuf, weights + (size_t)RR * DD * DD, roots + (size_t)DD * DD,
        biases + DD, hbuf, out, N);
    rgcn_count<<<(E + 255) / 256, 256, 0, stream>>>(dstI, etype, cnt, E, N);
    rgcn_scatter<<<scat_blocks, 256, 0, stream>>>(
        srcI, dstI, etype, cnt, hbuf, out, E, N);
}